// HomoGCNLayer_62045097558487
// MI455X (gfx1250) — compile-verified
//
#include <hip/hip_runtime.h>

#define LN_EPS  1e-5f
#define BB      8
#define NN      10000
#define CC      256
#define NCHUNK  80
#define CHUNK   125      // 80 * 125 == 10000
#define MPAD    16       // B padded to 16 rows for WMMA

typedef __attribute__((ext_vector_type(2))) float v2f;
typedef __attribute__((ext_vector_type(4))) float v4f;
typedef __attribute__((ext_vector_type(8))) float v8f;

// ---------------------------------------------------------------------------
// Kernel 1: partial column sums over node chunks.  grid = (B, NCHUNK), 256 thr.
// Thread c streams x[b, n0..n0+CHUNK, c]; fully coalesced 1 KB/row bursts.
// ---------------------------------------------------------------------------
__global__ __launch_bounds__(256) void colsum_partial_kernel(
    const float* __restrict__ x, float* __restrict__ partial) {
  const int b  = blockIdx.x;
  const int ch = blockIdx.y;
  const int c  = threadIdx.x;
  const float* xp = x + (((size_t)b * NN + (size_t)ch * CHUNK) * CC) + c;
  float s = 0.0f;
#pragma unroll 5
  for (int i = 0; i < CHUNK; ++i) s += xp[(size_t)i * CC];
  partial[((size_t)b * NCHUNK + ch) * CC + c] = s;
}

// ---------------------------------------------------------------------------
// Kernel 2: reduce partials -> mean[MPAD][CC]; rows 8..15 written as zero so
// the WMMA A-matrix (16 rows) is well defined.  Deterministic order.
// ---------------------------------------------------------------------------
__global__ __launch_bounds__(256) void colsum_final_kernel(
    const float* __restrict__ partial, float* __restrict__ meanp) {
  const int b = blockIdx.x;   // 0..15
  const int c = threadIdx.x;  // 0..255
  float s = 0.0f;
  if (b < BB) {
    const float* pp = partial + (size_t)b * NCHUNK * CC + c;
#pragma unroll 8
    for (int ch = 0; ch < NCHUNK; ++ch) s += pp[(size_t)ch * CC];
    s *= (1.0f / (float)NN);
  }
  meanp[b * CC + c] = s;
}

// ---------------------------------------------------------------------------
// Kernel 3: agg[16,256] = mean[16,256] @ W^T + bias  via V_WMMA_F32_16X16X4_F32
// One workgroup, 16 waves; wave w owns N-tile [16w, 16w+16), loops K by 4.
//   A (16x4 f32): lanes 0-15 -> M=lane, K={k0,k0+1}; lanes 16-31 -> K={k0+2,k0+3}
//   B (4x16):     B[k][n] = W[n][k]  (since agg = mean @ W^T)
//   C/D (16x16):  VGPR r, lanes 0-15 -> M=r, N=lane; lanes 16-31 -> M=8+r.
// ---------------------------------------------------------------------------
__global__ __launch_bounds__(512) void gemm_wmma_kernel(
    const float* __restrict__ meanp, const float* __restrict__ W,
    const float* __restrict__ bias, float* __restrict__ agg) {
  const int wave = threadIdx.x >> 5;   // 0..15 -> N tile index
  const int lane = threadIdx.x & 31;
  const int hh   = lane >> 4;          // half-wave: 0 or 1
  const int l    = lane & 15;
  const int n0   = wave * 16;

  v8f acc = {};
#pragma unroll 4
  for (int k0 = 0; k0 < CC; k0 += 4) {
    const int kk = k0 + 2 * hh;
    v2f a;   // A tile: mean[M=l][kk], mean[M=l][kk+1]
    a.x = meanp[l * CC + kk + 0];
    a.y = meanp[l * CC + kk + 1];
    v2f bm;  // B tile: W^T[kk][n0+l] = W[n0+l][kk]
    bm.x = W[(size_t)(n0 + l) * CC + kk + 0];
    bm.y = W[(size_t)(n0 + l) * CC + kk + 1];
    acc = __builtin_amdgcn_wmma_f32_16x16x4_f32(
        /*neg_a=*/false, a, /*neg_b=*/false, bm,
        /*c_mod=*/(short)0, acc, /*reuse_a=*/false, /*reuse_b=*/false);
  }

  const int n = n0 + l;
  const float bn = bias[n];
#pragma unroll
  for (int r = 0; r < 8; ++r) {
    const int m = hh * 8 + r;
    agg[m * CC + n] = acc[r] + bn;
  }
}

// ---------------------------------------------------------------------------
// Kernel 4: out = LayerNorm(x + agg[b]) * gamma + beta.
// One wave32 per row; 8 channels/lane as two b128 loads; shuffle reduction;
// non-temporal b128 stores so the 82 MB output stream doesn't evict x from L2.
// ---------------------------------------------------------------------------
__global__ __launch_bounds__(256) void fused_residual_ln_kernel(
    const float* __restrict__ x, const float* __restrict__ agg,
    const float* __restrict__ gamma, const float* __restrict__ beta,
    float* __restrict__ out) {
  const int lane = threadIdx.x & 31;
  const int wave = threadIdx.x >> 5;
  const int wavesTotal = gridDim.x * 8;
  const int c0 = lane * 8;

  const v4f g0 = *(const v4f*)(gamma + c0);
  const v4f g1 = *(const v4f*)(gamma + c0 + 4);
  const v4f e0 = *(const v4f*)(beta + c0);
  const v4f e1 = *(const v4f*)(beta + c0 + 4);

  for (int row = blockIdx.x * 8 + wave; row < BB * NN; row += wavesTotal) {
    const int b = row / NN;
    const float* xp = x + (size_t)row * CC + c0;
    const float* ap = agg + b * CC + c0;

    const v4f xv0 = *(const v4f*)(xp);
    const v4f xv1 = *(const v4f*)(xp + 4);
    const v4f av0 = *(const v4f*)(ap);
    const v4f av1 = *(const v4f*)(ap + 4);

    const v4f y0 = xv0 + av0;
    const v4f y1 = xv1 + av1;

    float s  = y0.x + y0.y + y0.z + y0.w + y1.x + y1.y + y1.z + y1.w;
    float s2 = y0.x * y0.x + y0.y * y0.y + y0.z * y0.z + y0.w * y0.w +
               y1.x * y1.x + y1.y * y1.y + y1.z * y1.z + y1.w * y1.w;
#pragma unroll
    for (int off = 16; off >= 1; off >>= 1) {
      s  += __shfl_xor(s, off, 32);
      s2 += __shfl_xor(s2, off, 32);
    }
    const float mu  = s * (1.0f / CC);
    const float var = fmaf(-mu, mu, s2 * (1.0f / CC));
    const float inv = rsqrtf(var + LN_EPS);

    v4f o0, o1;
    o0.x = (y0.x - mu) * inv * g0.x + e0.x;
    o0.y = (y0.y - mu) * inv * g0.y + e0.y;
    o0.z = (y0.z - mu) * inv * g0.z + e0.z;
    o0.w = (y0.w - mu) * inv * g0.w + e0.w;
    o1.x = (y1.x - mu) * inv * g1.x + e1.x;
    o1.y = (y1.y - mu) * inv * g1.y + e1.y;
    o1.z = (y1.z - mu) * inv * g1.z + e1.z;
    o1.w = (y1.w - mu) * inv * g1.w + e1.w;

    float* op = out + (size_t)row * CC + c0;
    __builtin_nontemporal_store(o0, (v4f*)(op));
    __builtin_nontemporal_store(o1, (v4f*)(op + 4));
  }
}

// ---------------------------------------------------------------------------
extern "C" void kernel_launch(void* const* d_in, const int* in_sizes, int n_in,
                              void* d_out, int out_size, void* d_ws, size_t ws_size,
                              hipStream_t stream) {
  (void)in_sizes; (void)n_in; (void)out_size; (void)ws_size;

  const float* x     = (const float*)d_in[0];
  const float* W     = (const float*)d_in[1];
  const float* bias  = (const float*)d_in[2];
  const float* gamma = (const float*)d_in[3];
  const float* beta  = (const float*)d_in[4];
  float* out = (float*)d_out;

  float* ws      = (float*)d_ws;
  float* partial = ws;                                   // 8*80*256   = 163840 f
  float* meanp   = partial + (size_t)BB * NCHUNK * CC;   // 16*256     =   4096 f
  float* agg     = meanp + MPAD * CC;                    // 16*256     =   4096 f

  colsum_partial_kernel<<<dim3(BB, NCHUNK), 256, 0, stream>>>(x, partial);
  colsum_final_kernel<<<MPAD, 256, 0, stream>>>(partial, meanp);
  gemm_wmma_kernel<<<1, 512, 0, stream>>>(meanp, W, bias, agg);
  fused_residual_ln_kernel<<<(BB * NN + 7) / 8, 256, 0, stream>>>(x, agg, gamma, beta, out);
}